// WaveletDWTLayer_78769700208778
// MI455X (gfx1250) — compile-verified
//
#include <hip/hip_runtime.h>
#include <stdint.h>

// db4 DWT (pywt mode='symmetric') on MI455X / gfx1250.
//
// Memory-bound: 268 MB of HBM traffic vs ~0.5 GFLOP -> ~11.5 us floor at
// 23.3 TB/s. CDNA5 path used: GLOBAL_LOAD_ASYNC_TO_LDS_B128 (ASYNCcnt) to
// stream the input tile into LDS with no VGPR round trip, then f32 VALU FMAs
// out of LDS (each input float feeds 8 overlapping stride-2 taps).
// WMMA deliberately NOT used: only 2 output channels -> a 16x16 tile would be
// 87% idle, and arithmetic intensity (~2 FLOP/B) is ~50x below any matrix-op
// ridge point; the tensor path buys nothing on a streaming op.

namespace {

constexpr int BROWS   = 512;
constexpr int NIN     = 65536;
constexpr int MOUT    = 32771;            // (NIN + 8 - 1) / 2
constexpr int TILE_J  = 1024;             // outputs (per channel) per block
constexpr int TILE_IN = 2 * TILE_J;       // 2048 bulk input floats per tile
constexpr int NTILES  = NIN / TILE_IN;    // 32 tiles -> j in [0, 32768)
constexpr int JEDGE   = NTILES * TILE_J;  // 32768; j = 32768..32770 -> edge kernel
constexpr int THREADS = 256;

// Low 32 bits of a flat LDS-aperture address == LDS byte offset
__device__ __forceinline__ uint32_t lds_off(const void* p) {
  return (uint32_t)(uintptr_t)p;
}

// Async DMA: 16 bytes global -> LDS per lane. Tracked by ASYNCcnt.
__device__ __forceinline__ void async_copy_b128(uint32_t lds_byte, const float* g) {
  asm volatile("global_load_async_to_lds_b128 %0, %1, off"
               :: "v"(lds_byte), "v"(g)
               : "memory");
}

__device__ __forceinline__ void wait_asynccnt0() {
#if __has_builtin(__builtin_amdgcn_s_wait_asynccnt)
  __builtin_amdgcn_s_wait_asynccnt(0);
#else
  asm volatile("s_wait_asynccnt 0" ::: "memory");
#endif
}

// Interior tiles: grid = (32 tiles, 512 rows). Each block: 1024 j's of one row.
// s_x[idx] = X(g0 + idx - 8); bulk s_x[8..2055] <- x[g0 .. g0+2047] (async DMA),
// left halo s_x[2..7] <- X(g0-6 .. g0-1) (reflected only for tile 0).
__global__ __launch_bounds__(THREADS)
void dwt_db4_main(const float* __restrict__ x1, const float* __restrict__ w,
                  float* __restrict__ outA, float* __restrict__ outD) {
  __shared__ __align__(16) float s_x[8 + TILE_IN];

  const int  tile = blockIdx.x;
  const int  row  = blockIdx.y;
  const int  tid  = threadIdx.x;
  const long g0   = (long)tile * TILE_IN;
  const float* xrow = x1 + (long)row * NIN;

  // Stage 2048 floats: 512 x b128 transactions, 2 per lane, fully aligned.
#pragma unroll
  for (int i = 0; i < 2; ++i) {
    const int f = 4 * tid + 1024 * i;
    async_copy_b128(lds_off(&s_x[8 + f]), xrow + g0 + f);
  }
  // 6-float left halo with symmetric reflection at the row start.
  if (tid < 6) {
    const long t = g0 - 6 + tid;
    s_x[2 + tid] = (t >= 0) ? xrow[t] : xrow[-1 - t];
  }
  wait_asynccnt0();
  __syncthreads();

  const float w0 = w[0], w1 = w[1], w2 = w[2], w3 = w[3];
  const float w4 = w[4], w5 = w[5], w6 = w[6], w7 = w[7];

#pragma unroll
  for (int i = 0; i < 4; ++i) {
    const int n = tid + THREADS * i;          // local output index, stores coalesce
    const float* p = &s_x[2 * n + 2];         // = X(2j - 6) ...
    const float a0 = p[0], a1 = p[1], a2 = p[2], a3 = p[3];
    const float a4 = p[4], a5 = p[5], a6 = p[6], a7 = p[7];
    // cA[j] = sum_k rec_lo[k] * X(2j+k-6)
    const float cA = ((w0 * a0 + w1 * a1) + (w2 * a2 + w3 * a3)) +
                     ((w4 * a4 + w5 * a5) + (w6 * a6 + w7 * a7));
    // cD[j] = sum_k (-1)^k * rec_lo[7-k] * X(2j+k-6)
    const float cD = ((w7 * a0 - w6 * a1) + (w5 * a2 - w4 * a3)) +
                     ((w3 * a4 - w2 * a5) + (w1 * a6 - w0 * a7));
    const long j = (long)tile * TILE_J + n;
    const long o = (long)row * MOUT + j;
    __builtin_nontemporal_store(cA, &outA[o]);  // outputs never re-read
    __builtin_nontemporal_store(cD, &outD[o]);
  }
}

// Right-edge outputs (3 per row, symmetric reflection) + x3 passthrough.
__global__ __launch_bounds__(256)
void dwt_db4_edge_and_copy(const float* __restrict__ x1, const float* __restrict__ w,
                           const float* __restrict__ x3,
                           float* __restrict__ outA, float* __restrict__ outD,
                           float* __restrict__ outX3) {
  const int id = blockIdx.x * blockDim.x + threadIdx.x;
  const int n_edge = BROWS * 3;
  if (id < n_edge) {
    const int row = id / 3;
    const int jj  = JEDGE + (id % 3);
    const float* xrow = x1 + (long)row * NIN;
    float cA = 0.0f, cD = 0.0f;
#pragma unroll
    for (int k = 0; k < 8; ++k) {
      const long t = 2L * jj + k - 6;                       // always >= 0 here
      const float v  = (t < NIN) ? xrow[t] : xrow[2L * NIN - 1 - t];
      const float hk = w[k];
      const float gk = (k & 1) ? -w[7 - k] : w[7 - k];
      cA += hk * v;
      cD += gk * v;
    }
    const long o = (long)row * MOUT + jj;
    outA[o] = cA;
    outD[o] = cD;
  } else {
    const int m = id - n_edge;
    if (m < BROWS * 8) outX3[m] = x3[m];
  }
}

}  // namespace

extern "C" void kernel_launch(void* const* d_in, const int* in_sizes, int n_in,
                              void* d_out, int out_size, void* d_ws, size_t ws_size,
                              hipStream_t stream) {
  // setup_inputs order: x1 (512*65536 f32), x2 (unused), x3 (512*8 f32), weights (8 f32)
  const float* x1 = (const float*)d_in[0];
  const float* x3 = (const float*)d_in[2];
  const float* w  = (const float*)d_in[3];

  float* out   = (float*)d_out;
  float* outA  = out;                                 // (512, 32771)
  float* outD  = out + (size_t)BROWS * MOUT;          // (512, 1, 32771)
  float* outX3 = out + 2 * (size_t)BROWS * MOUT;      // (512, 1, 8)

  dim3 grid(NTILES, BROWS);
  dwt_db4_main<<<grid, THREADS, 0, stream>>>(x1, w, outA, outD);

  const int tail = BROWS * 3 + BROWS * 8;             // 1536 edge outputs + 4096 copies
  dwt_db4_edge_and_copy<<<(tail + 255) / 256, 256, 0, stream>>>(x1, w, x3, outA, outD, outX3);
}